// Model04_54331336294860
// MI455X (gfx1250) — compile-verified
//
#include <hip/hip_runtime.h>
#include <hip/hip_bf16.h>

#define BATCH  4096
#define TSEQ   512
#define DIN    16
#define NLAYER 6

typedef float v2f __attribute__((ext_vector_type(2)));
typedef float v8f __attribute__((ext_vector_type(8)));

__device__ __forceinline__ float sigf(float x) {
    return 1.0f / (1.0f + __expf(-x));
}
__device__ __forceinline__ float tanh_fast(float x) {
    float ax = fabsf(x);
    float e  = __expf(-2.0f * ax);
    float r  = (1.0f - e) / (1.0f + e);
    return copysignf(r, x);
}

// ---------------------------------------------------------------------------
// Kernel 1: layer-0 input projection via V_WMMA_F32_16X16X4_F32.
// xproj[t][b][g] = sum_k x[b][t][k] * w_ih0[g][k] + (b_ih[0][g] + b_hh[0][g])
//
// Tile = 16 consecutive timesteps of ONE batch element (WMMA rows = t).
// The wave's 4 A-operand loads then cover a fully contiguous 1 KB region of
// x (the dominant 128 MB stream -> clean first-touch DRAM efficiency).
// K=16 is covered by 4 chained fp32 WMMAs (K=4 each). Accumulator is
// initialized with the per-gate bias. Output layout [T][B][4] fp32.
// ---------------------------------------------------------------------------
__global__ __launch_bounds__(256) void wmma_xproj_kernel(
    const float* __restrict__ x,      // [B][T][16]
    const float* __restrict__ w_ih0,  // [4][16]
    const float* __restrict__ b_ih,   // [6][4]
    const float* __restrict__ b_hh,   // [6][4]
    float* __restrict__ xproj)        // [T][B][4]
{
    const int lane = threadIdx.x & 31;
    const int wave = threadIdx.x >> 5;
    const int tile = blockIdx.x * 8 + wave;   // 4096 * 32 = 131072 tiles
    const int b    = tile >> 5;               // batch element
    const int t0   = (tile & 31) << 4;        // first of 16 timesteps

    const int m    = lane & 15;        // A-matrix row (t offset within tile)
    const int half = lane >> 4;        // K sub-pair selector per ISA layout
    const int n    = lane & 15;        // B/C column (gate index when < 4)
    const int nc   = (n < 4) ? n : 3;  // clamp to stay in-bounds

    // Accumulator pre-loaded with per-gate bias (constant down the M dim).
    const float biasn = (n < 4) ? (b_ih[nc] + b_hh[nc]) : 0.0f;
    v8f acc;
#pragma unroll
    for (int j = 0; j < 8; ++j) acc[j] = biasn;

    // Row M = m corresponds to x[b][t0 + m][*]; contiguous rows 64 B apart.
    const float* xr = x + ((size_t)b * TSEQ + t0 + m) * DIN;

#pragma unroll
    for (int c = 0; c < 4; ++c) {
        const int kg = 4 * c + 2 * half;
        // A 16x4 f32: lane<16 holds K={4c,4c+1}, lane>=16 holds K={4c+2,4c+3}
        v2f a = *(const v2f*)(xr + kg);
        // B 4x16 f32: VGPR0 rows K={4c,4c+2}, VGPR1 rows K={4c+1,4c+3};
        // only columns n<4 carry gate weights, the rest are zeroed.
        float w0 = w_ih0[nc * DIN + kg];
        float w1 = w_ih0[nc * DIN + kg + 1];
        v2f bm;
        bm.x = (n < 4) ? w0 : 0.0f;
        bm.y = (n < 4) ? w1 : 0.0f;
        acc = __builtin_amdgcn_wmma_f32_16x16x4_f32(
            /*neg_a=*/false, a, /*neg_b=*/false, bm,
            /*c_mod=*/(short)0, acc, /*reuse_a=*/false, /*reuse_b=*/false);
    }

    // C/D layout: VGPR j holds (M=j, N=lane) for lanes 0-15 and (M=j+8,
    // N=lane-16) for lanes 16-31. Row M -> t = t0 + j + 8*half.
    if (n < 4) {
        float* op = xproj + ((size_t)(t0 + (half << 3)) * BATCH + b) * 4 + n;
#pragma unroll
        for (int j = 0; j < 8; ++j) op[(size_t)j * BATCH * 4] = acc[j];
    }
}

// ---------------------------------------------------------------------------
// Kernel 2: wave-per-layer pipelined LSTM scan.
// Workgroup = 192 threads = 6 waves; wave l owns layer l for 32 batch
// elements (one per lane). Skewed schedule: at step s, wave l processes
// t = s - l. Hidden states cross layers through a double-buffered LDS
// array: step s writes hbuf[s&1], reads hbuf[(s-1)&1], so a single
// __syncthreads() per step is sufficient. This cuts the sequential
// per-wave instruction count ~6x vs a per-thread scan and runs 768 waves.
// ---------------------------------------------------------------------------
__global__ __launch_bounds__(192) void lstm_pipeline_kernel(
    const float* __restrict__ xproj,      // [T][B][4], layer-0 bias included
    const float* __restrict__ w_ih_rest,  // [5][4][1]
    const float* __restrict__ w_hh,       // [6][4][1]
    const float* __restrict__ b_ih,       // [6][4]
    const float* __restrict__ b_hh,       // [6][4]
    float* __restrict__ out)              // [B]
{
    __shared__ float hbuf[2][NLAYER][32];

    const int lane = threadIdx.x & 31;
    const int l    = threadIdx.x >> 5;        // layer handled by this wave
    const int b    = blockIdx.x * 32 + lane;  // batch element of this lane

    // Per-layer parameters (uniform within the wave).
    float wihk[4], whhk[4], biask[4];
    if (l == 0) {
#pragma unroll
        for (int k = 0; k < 4; ++k) {
            wihk[k] = 0.0f; biask[k] = 0.0f; whhk[k] = w_hh[k];
        }
    } else {
#pragma unroll
        for (int k = 0; k < 4; ++k) {
            wihk[k]  = w_ih_rest[(l - 1) * 4 + k];
            whhk[k]  = w_hh[l * 4 + k];
            biask[k] = b_ih[l * 4 + k] + b_hh[l * 4 + k];
        }
    }

    float h = 0.0f, cst = 0.0f;
    const float4* xp4 = (const float4*)xproj;

    for (int s = 0; s < TSEQ + NLAYER - 1; ++s) {
        const int tl = s - l;                  // uniform per wave
        if (tl >= 0 && tl < TSEQ) {
            float g0, g1, g2, g3;
            if (l == 0) {
                float4 xg = xp4[(size_t)tl * BATCH + b];  // coalesced 512 B
                g0 = xg.x + h * whhk[0];
                g1 = xg.y + h * whhk[1];
                g2 = xg.z + h * whhk[2];
                g3 = xg.w + h * whhk[3];
            } else {
                // h of layer l-1 at time tl, written by wave l-1 last step.
                const float hin = hbuf[(s + 1) & 1][l - 1][lane];
                g0 = fmaf(hin, wihk[0], biask[0]) + h * whhk[0];
                g1 = fmaf(hin, wihk[1], biask[1]) + h * whhk[1];
                g2 = fmaf(hin, wihk[2], biask[2]) + h * whhk[2];
                g3 = fmaf(hin, wihk[3], biask[3]) + h * whhk[3];
            }
            const float gi = sigf(g0);
            const float gf = sigf(g1);
            const float gg = tanh_fast(g2);
            const float go = sigf(g3);
            cst = gf * cst + gi * gg;
            h   = go * tanh_fast(cst);
            hbuf[s & 1][l][lane] = h;
        }
        __syncthreads();   // separates step s writes from step s+1 reads
    }

    if (l == NLAYER - 1) out[b] = h;  // y[:, -1, 0] with H == 1
}

// ---------------------------------------------------------------------------
// Fallback (only if ws_size < 32 MB): fully fused per-thread scan with the
// layer-0 projection computed inline from x. No scratch required.
// ---------------------------------------------------------------------------
__global__ __launch_bounds__(256) void lstm_fused_kernel(
    const float* __restrict__ x,
    const float* __restrict__ w_ih0,
    const float* __restrict__ w_ih_rest,
    const float* __restrict__ w_hh,
    const float* __restrict__ b_ih,
    const float* __restrict__ b_hh,
    float* __restrict__ out)
{
    const int b = blockIdx.x * blockDim.x + threadIdx.x;

    float w0[4][DIN];
#pragma unroll
    for (int g = 0; g < 4; ++g)
#pragma unroll
        for (int k = 0; k < DIN; ++k) w0[g][k] = w_ih0[g * DIN + k];

    float wih[NLAYER][4], whh[NLAYER][4], bias[NLAYER][4];
#pragma unroll
    for (int k = 0; k < 4; ++k) {
        wih[0][k] = 0.0f; bias[0][k] = b_ih[k] + b_hh[k]; whh[0][k] = w_hh[k];
    }
#pragma unroll
    for (int l = 1; l < NLAYER; ++l)
#pragma unroll
        for (int k = 0; k < 4; ++k) {
            wih[l][k]  = w_ih_rest[(l - 1) * 4 + k];
            whh[l][k]  = w_hh[l * 4 + k];
            bias[l][k] = b_ih[l * 4 + k] + b_hh[l * 4 + k];
        }

    float h[NLAYER], cc[NLAYER];
#pragma unroll
    for (int l = 0; l < NLAYER; ++l) { h[l] = 0.0f; cc[l] = 0.0f; }

    for (int s = 0; s < TSEQ + NLAYER - 1; ++s) {
#pragma unroll
        for (int l = NLAYER - 1; l >= 0; --l) {
            const int tl = s - l;
            if (tl >= 0 && tl < TSEQ) {
                float g0, g1, g2, g3;
                if (l == 0) {
                    const float* xr = x + ((size_t)b * TSEQ + tl) * DIN;
                    float a0 = bias[0][0], a1 = bias[0][1];
                    float a2 = bias[0][2], a3 = bias[0][3];
#pragma unroll
                    for (int k = 0; k < DIN; ++k) {
                        const float xv = xr[k];
                        a0 = fmaf(xv, w0[0][k], a0);
                        a1 = fmaf(xv, w0[1][k], a1);
                        a2 = fmaf(xv, w0[2][k], a2);
                        a3 = fmaf(xv, w0[3][k], a3);
                    }
                    g0 = a0 + h[0] * whh[0][0];
                    g1 = a1 + h[0] * whh[0][1];
                    g2 = a2 + h[0] * whh[0][2];
                    g3 = a3 + h[0] * whh[0][3];
                } else {
                    const float in = h[l - 1];
                    g0 = fmaf(in, wih[l][0], bias[l][0]) + h[l] * whh[l][0];
                    g1 = fmaf(in, wih[l][1], bias[l][1]) + h[l] * whh[l][1];
                    g2 = fmaf(in, wih[l][2], bias[l][2]) + h[l] * whh[l][2];
                    g3 = fmaf(in, wih[l][3], bias[l][3]) + h[l] * whh[l][3];
                }
                const float gi = sigf(g0);
                const float gf = sigf(g1);
                const float gg = tanh_fast(g2);
                const float go = sigf(g3);
                cc[l] = gf * cc[l] + gi * gg;
                h[l]  = go * tanh_fast(cc[l]);
            }
        }
    }
    out[b] = h[NLAYER - 1];
}

extern "C" void kernel_launch(void* const* d_in, const int* in_sizes, int n_in,
                              void* d_out, int out_size, void* d_ws, size_t ws_size,
                              hipStream_t stream) {
    const float* x         = (const float*)d_in[0];
    const float* w_ih0     = (const float*)d_in[1];
    const float* w_ih_rest = (const float*)d_in[2];
    const float* w_hh      = (const float*)d_in[3];
    const float* b_ih      = (const float*)d_in[4];
    const float* b_hh      = (const float*)d_in[5];
    float* out             = (float*)d_out;

    const size_t need = (size_t)TSEQ * BATCH * 4 * sizeof(float);  // 32 MiB
    if (ws_size >= need) {
        float* xproj = (float*)d_ws;
        // 4096 * 32 = 131072 wave-tiles, 8 waves (256 thr) per block
        hipLaunchKernelGGL(wmma_xproj_kernel, dim3(131072 / 8), dim3(256), 0, stream,
                           x, w_ih0, b_ih, b_hh, xproj);
        // 128 workgroups x 6 waves (one wave per LSTM layer)
        hipLaunchKernelGGL(lstm_pipeline_kernel, dim3(BATCH / 32), dim3(192), 0, stream,
                           xproj, w_ih_rest, w_hh, b_ih, b_hh, out);
    } else {
        hipLaunchKernelGGL(lstm_fused_kernel, dim3(BATCH / 256), dim3(256), 0, stream,
                           x, w_ih0, w_ih_rest, w_hh, b_ih, b_hh, out);
    }
}